// GraphLearning_15908558864644
// MI455X (gfx1250) — compile-verified
//
#include <hip/hip_runtime.h>
#include <math.h>

// ---------------- problem constants (match reference) ----------------
#define NN   50000
#define EE   800000
#define INF_ 512
#define FF   32
#define HH   4
#define CC   32
#define HCC  128
#define EDD  4
#define E2E  850000   // EE + NN (self loops)
#define LL   2

typedef float v2f __attribute__((ext_vector_type(2)));
typedef float v8f __attribute__((ext_vector_type(8)));

// ---------------- helpers ----------------
__device__ __forceinline__ unsigned fkey(float f) {
    unsigned u = __float_as_uint(f);
    return (u & 0x80000000u) ? ~u : (u | 0x80000000u);
}
__device__ __forceinline__ float funkey(unsigned u) {
    return __uint_as_float((u & 0x80000000u) ? (u & 0x7FFFFFFFu) : ~u);
}
__device__ __forceinline__ float leakyf(float x, float s) {
    return x >= 0.0f ? x : s * x;
}

// ---------------- generic f32 WMMA GEMM: C = A(MxK) @ W^T (+bias) ----------------
// W is (Nout x K) row-major. One wave computes a 16 x (16*CT) output strip with
// V_WMMA_F32_16X16X4_F32.  A fragments are register double-buffered (software
// pipeline) so loads for step k+4 are in flight during step k's WMMAs.
// USE_LDS: block cooperatively stages W into LDS with a (KT+4)-float padded row
// stride -> (KT/4+1) odd -> the 32 ds_load_b64 lanes hit all 64 banks, no conflicts.
template <int CT, int KT, bool USE_LDS>
__global__ __launch_bounds__(256)
void gemm_wmma_f32(const float* __restrict__ A, const float* __restrict__ W,
                   const float* __restrict__ bias, float* __restrict__ Cout,
                   int M, int Nout) {
    extern __shared__ float sW[];
    const int tid  = (int)threadIdx.x;
    const int lane = tid & 31;
    const int wave = (int)((blockIdx.x * blockDim.x + tid) >> 5);
    const int mtiles  = M >> 4;
    const int ngroups = Nout / (16 * CT);
    const int stride  = KT + 4;

    if (USE_LDS) {
        const int total = Nout * KT;           // KT is a power of two
        for (int i = tid; i < total; i += 256) {
            const int c  = i / KT;
            const int kk = i - c * KT;
            sW[c * stride + kk] = W[i];
        }
        __syncthreads();
    }
    if (wave >= mtiles * ngroups) return;

    const int mtile = wave / ngroups;
    const int ng    = wave % ngroups;
    const int row0  = mtile * 16;
    const int col0  = ng * 16 * CT;
    const int hi = lane >> 4;      // K sub-pair (0: k,k+1   1: k+2,k+3)
    const int mi = lane & 15;      // row (A) / col (B/C) within tile

    v8f acc[CT] = {};
    const float* Arow = A + (size_t)(row0 + mi) * KT + 2 * hi;

    v2f a_cur;
    a_cur.x = Arow[0];
    a_cur.y = Arow[1];
    v2f b_cur[CT];
    if (!USE_LDS) {
#pragma unroll
        for (int t = 0; t < CT; ++t) {
            const float* Wc = W + (size_t)(col0 + t * 16 + mi) * KT + 2 * hi;
            b_cur[t].x = Wc[0];
            b_cur[t].y = Wc[1];
        }
    }

    for (int k = 0; k < KT; k += 4) {
        v2f a_nxt = {};
        v2f b_nxt[CT];
        if (k + 4 < KT) {
            a_nxt.x = Arow[k + 4];
            a_nxt.y = Arow[k + 5];
            if (!USE_LDS) {
#pragma unroll
                for (int t = 0; t < CT; ++t) {
                    const float* Wc = W + (size_t)(col0 + t * 16 + mi) * KT + 2 * hi;
                    b_nxt[t].x = Wc[k + 4];
                    b_nxt[t].y = Wc[k + 5];
                }
                if ((k & 31) == 0)  // stream-prefetch A ~0.5KB ahead
                    __builtin_prefetch(Arow + k + 132, 0, 1);
            }
        }
#pragma unroll
        for (int t = 0; t < CT; ++t) {
            v2f b;
            if (USE_LDS) {
                const float* Sc = sW + (size_t)(col0 + t * 16 + mi) * stride + 2 * hi;
                b.x = Sc[k];
                b.y = Sc[k + 1];
            } else {
                b = b_cur[t];
            }
            acc[t] = __builtin_amdgcn_wmma_f32_16x16x4_f32(
                false, a_cur, false, b, (short)0, acc[t], false, false);
        }
        a_cur = a_nxt;
        if (!USE_LDS) {
#pragma unroll
            for (int t = 0; t < CT; ++t) b_cur[t] = b_nxt[t];
        }
    }

#pragma unroll
    for (int t = 0; t < CT; ++t) {
        const int col = col0 + t * 16 + mi;
        const float bb = bias ? bias[col] : 0.0f;
#pragma unroll
        for (int v = 0; v < 8; ++v) {
            const int r = row0 + v + 8 * hi;   // C layout: VGPR v -> M=v (+8 hi half)
            Cout[(size_t)r * Nout + col] = acc[t][v] + bb;
        }
    }
}

// ---------------- per-row LayerNorm(32) + leaky + JK-max ----------------
__global__ __launch_bounds__(256)
void ln_rows_kernel(const float* __restrict__ Z, const float* __restrict__ w,
                    const float* __restrict__ b, float* __restrict__ Hout,
                    float* __restrict__ jk, int mode, int M) {
    const int row  = (int)((blockIdx.x * blockDim.x + threadIdx.x) >> 5);
    const int lane = (int)(threadIdx.x & 31);
    if (row >= M) return;
    float v = Z[(size_t)row * FF + lane];
    float s = v;
#pragma unroll
    for (int off = 16; off >= 1; off >>= 1) s += __shfl_xor(s, off);
    const float mu = s * (1.0f / 32.0f);
    const float d = v - mu;
    float q = d * d;
#pragma unroll
    for (int off = 16; off >= 1; off >>= 1) q += __shfl_xor(q, off);
    const float var = q * (1.0f / 32.0f);
    float y = w[lane] * d * rsqrtf(var + 1e-5f) + b[lane];
    y = leakyf(y, 0.01f);
    Hout[(size_t)row * FF + lane] = y;
    float* o = jk + (size_t)row * FF + lane;
    *o = (mode == 0) ? y : fmaxf(*o, y);
}

// ---------------- edge pass 1: GATv2 scores + segment max ----------------
// one wave per edge; lane owns 4 consecutive channels (float4), head = lane>>3
__global__ __launch_bounds__(256)
void edge_score_kernel(const int* __restrict__ src, const int* __restrict__ dst,
                       const float* __restrict__ ea, const float* __restrict__ We,
                       const float* __restrict__ xl, const float* __restrict__ xr,
                       const float* __restrict__ att, float* __restrict__ score,
                       unsigned* __restrict__ smax) {
    const int e    = (int)((blockIdx.x * blockDim.x + threadIdx.x) >> 5);
    const int lane = (int)(threadIdx.x & 31);
    if (e >= E2E) return;
    int sN, dN;
    float4 ev;
    if (e < EE) {
        sN = src[e]; dN = dst[e];
        ev = ((const float4*)ea)[e];
    } else {  // self loop, edge_attr fill = 0
        sN = dN = e - EE;
        ev = make_float4(0.0f, 0.0f, 0.0f, 0.0f);
    }
    const float4 xlv = ((const float4*)xl)[(size_t)sN * (HCC / 4) + lane];
    const float4 xrv = ((const float4*)xr)[(size_t)dN * (HCC / 4) + lane];
    const float4 atv = ((const float4*)att)[lane];
    const float4* We4 = (const float4*)We;

    const float xa[4] = {xlv.x, xlv.y, xlv.z, xlv.w};
    const float xb[4] = {xrv.x, xrv.y, xrv.z, xrv.w};
    const float at[4] = {atv.x, atv.y, atv.z, atv.w};
    float sc = 0.0f;
#pragma unroll
    for (int k = 0; k < 4; ++k) {
        const float4 wr = We4[lane * 4 + k];
        const float eev = wr.x * ev.x + wr.y * ev.y + wr.z * ev.z + wr.w * ev.w;
        float m = xa[k] + xb[k] + eev;
        m = leakyf(m, 0.2f);
        sc += at[k] * m;
    }
    // reduce over the 8 lanes of this head
    sc += __shfl_xor(sc, 4);
    sc += __shfl_xor(sc, 2);
    sc += __shfl_xor(sc, 1);
    if ((lane & 7) == 0) {
        const int h = lane >> 3;
        score[(size_t)e * HH + h] = sc;
        atomicMax(&smax[(size_t)dN * HH + h], fkey(sc));
    }
}

// ---------------- edge pass 2: softmax denominator ----------------
__global__ __launch_bounds__(256)
void edge_den_kernel(const int* __restrict__ dst, const float* __restrict__ score,
                     const unsigned* __restrict__ smax, float* __restrict__ den) {
    const int t = (int)(blockIdx.x * blockDim.x + threadIdx.x);
    if (t >= E2E * HH) return;
    const int e = t >> 2;
    const int h = t & 3;
    const int dN = (e < EE) ? dst[e] : (e - EE);
    const float ex = expf(score[t] - funkey(smax[(size_t)dN * HH + h]));
    atomicAdd(&den[(size_t)dN * HH + h], ex);
}

// ---------------- edge pass 3: weighted scatter aggregation ----------------
__global__ __launch_bounds__(256)
void edge_agg_kernel(const int* __restrict__ src, const int* __restrict__ dst,
                     const float* __restrict__ score, const unsigned* __restrict__ smax,
                     const float* __restrict__ den, const float* __restrict__ xl,
                     float* __restrict__ agg) {
    const int e    = (int)((blockIdx.x * blockDim.x + threadIdx.x) >> 5);
    const int lane = (int)(threadIdx.x & 31);
    if (e >= E2E) return;
    int sN, dN;
    if (e < EE) { sN = src[e]; dN = dst[e]; }
    else        { sN = dN = e - EE; }
    const int h = lane >> 3;
    const float mx = funkey(smax[(size_t)dN * HH + h]);
    const float dn = den[(size_t)dN * HH + h];
    const float alpha = expf(score[(size_t)e * HH + h] - mx) / (dn + 1e-16f);
    const float4 xv = ((const float4*)xl)[(size_t)sN * (HCC / 4) + lane];
    float* ao = agg + (size_t)dN * HCC + lane * 4;
    atomicAdd(ao + 0, xv.x * alpha);
    atomicAdd(ao + 1, xv.y * alpha);
    atomicAdd(ao + 2, xv.z * alpha);
    atomicAdd(ao + 3, xv.w * alpha);
}

// ---------------- conv bias add + global sum/sumsq (pyg graph-LN stats) ----------------
__global__ __launch_bounds__(256)
void bias_stats_kernel(float* __restrict__ agg, const float* __restrict__ cb,
                       double* __restrict__ red, int total) {
    __shared__ double ss[256];
    __shared__ double sq[256];
    const int idx = (int)(blockIdx.x * blockDim.x + threadIdx.x);
    double s = 0.0, q = 0.0;
    if (idx < total) {
        const float y = agg[idx] + cb[idx & (HCC - 1)];
        agg[idx] = y;
        s = (double)y;
        q = (double)y * (double)y;
    }
    ss[threadIdx.x] = s;
    sq[threadIdx.x] = q;
    __syncthreads();
    for (int o = 128; o > 0; o >>= 1) {
        if ((int)threadIdx.x < o) {
            ss[threadIdx.x] += ss[threadIdx.x + o];
            sq[threadIdx.x] += sq[threadIdx.x + o];
        }
        __syncthreads();
    }
    if (threadIdx.x == 0) {
        atomicAdd(&red[0], ss[0]);
        atomicAdd(&red[1], sq[0]);
    }
}

// ---------------- apply pyg graph-LN + leaky 0.01, in place ----------------
__global__ __launch_bounds__(256)
void pln_apply_kernel(float* __restrict__ agg, const float* __restrict__ w,
                      const float* __restrict__ b, const double* __restrict__ red,
                      int total) {
    const int idx = (int)(blockIdx.x * blockDim.x + threadIdx.x);
    if (idx >= total) return;
    const double mu = red[0] / (double)total;
    double var = red[1] / (double)total - mu * mu;
    if (var < 0.0) var = 0.0;
    const float sd = sqrtf((float)var);
    const int c = idx & (HCC - 1);
    float y = w[c] * (float)((agg[idx] - mu) / (double)(sd + 1e-5f)) + b[c];
    agg[idx] = leakyf(y, 0.01f);
}

// ---------------- host driver ----------------
extern "C" void kernel_launch(void* const* d_in, const int* in_sizes, int n_in,
                              void* d_out, int out_size, void* d_ws, size_t ws_size,
                              hipStream_t stream) {
    const float* x          = (const float*)d_in[0];
    const int*   edge_index = (const int*)  d_in[1];
    const float* edge_attr  = (const float*)d_in[2];
    const float* embed_W    = (const float*)d_in[3];
    const float* embed_b    = (const float*)d_in[4];
    const float* embed_lnw  = (const float*)d_in[5];
    const float* embed_lnb  = (const float*)d_in[6];
    const float* Wl         = (const float*)d_in[7];
    const float* bl         = (const float*)d_in[8];
    const float* Wr         = (const float*)d_in[9];
    const float* br         = (const float*)d_in[10];
    const float* We         = (const float*)d_in[11];
    const float* att        = (const float*)d_in[12];
    const float* conv_b     = (const float*)d_in[13];
    const float* pln_w      = (const float*)d_in[14];
    const float* pln_b      = (const float*)d_in[15];
    const float* proj_W     = (const float*)d_in[16];
    const float* proj_b     = (const float*)d_in[17];
    const float* ln2_w      = (const float*)d_in[18];
    const float* ln2_b      = (const float*)d_in[19];

    const int* srcE = edge_index;        // edge_index[0,:]
    const int* dstE = edge_index + EE;   // edge_index[1,:]

    // ---- workspace carve-out ----
    size_t off = 0;
    auto carve = [&](size_t nbytes) -> char* {
        char* p = (char*)d_ws + off;
        off += (nbytes + 255) & ~(size_t)255;
        return p;
    };
    float*    h     = (float*)   carve((size_t)NN * FF  * 4);
    float*    raw   = (float*)   carve((size_t)NN * FF  * 4);
    float*    xl    = (float*)   carve((size_t)NN * HCC * 4);
    float*    xr    = (float*)   carve((size_t)NN * HCC * 4);
    float*    agg   = (float*)   carve((size_t)NN * HCC * 4);
    float*    score = (float*)   carve((size_t)E2E * HH * 4);
    unsigned* smax  = (unsigned*)carve((size_t)NN * HH  * 4);
    float*    den   = (float*)   carve((size_t)NN * HH  * 4);
    double*   red   = (double*)  carve(2 * sizeof(double));
    (void)ws_size; (void)in_sizes; (void)n_in; (void)out_size;

    float* jk = (float*)d_out;

    const int mtiles = NN / 16;                       // 3125
    const dim3 blk(256);
    const dim3 gGemm((mtiles * 32 + 255) / 256);      // 1 wave / 16-row strip
    const dim3 gRows((NN * 32 + 255) / 256);          // 1 wave / row
    const dim3 gEdgeW(((size_t)E2E * 32 + 255) / 256);
    const dim3 gEdgeT(((size_t)E2E * HH + 255) / 256);
    const dim3 gElem(((size_t)NN * HCC + 255) / 256);
    const int totalHC = NN * HCC;

    const size_t shLin  = (size_t)HCC * (FF + 4) * 4;   // 18432 B
    const size_t shProj = (size_t)FF * (HCC + 4) * 4;   // 16896 B

    // ---- node embed: h0 = leaky(LN(x @ embed_W^T + b)) ----
    gemm_wmma_f32<2, INF_, false><<<gGemm, blk, 0, stream>>>(x, embed_W, embed_b, raw, NN, FF);
    ln_rows_kernel<<<gRows, blk, 0, stream>>>(raw, embed_lnw, embed_lnb, h, jk, 0, NN);

    for (int l = 0; l < LL; ++l) {
        const float* Wl_l = Wl + (size_t)l * HCC * FF;
        const float* bl_l = bl + (size_t)l * HCC;
        const float* Wr_l = Wr + (size_t)l * HCC * FF;
        const float* br_l = br + (size_t)l * HCC;
        const float* We_l = We + (size_t)l * HCC * EDD;
        const float* at_l = att + (size_t)l * HH * CC;
        const float* cb_l = conv_b + (size_t)l * HCC;
        const float* pw_l = pln_w + (size_t)l * HCC;
        const float* pb_l = pln_b + (size_t)l * HCC;
        const float* pW_l = proj_W + (size_t)l * FF * HCC;
        const float* pB_l = proj_b + (size_t)l * FF;
        const float* l2w  = ln2_w + (size_t)l * FF;
        const float* l2b  = ln2_b + (size_t)l * FF;

        // lin_l / lin_r  (N x 32 @ 32 x 128), W staged in LDS
        gemm_wmma_f32<8, FF, true><<<gGemm, blk, shLin, stream>>>(h, Wl_l, bl_l, xl, NN, HCC);
        gemm_wmma_f32<8, FF, true><<<gGemm, blk, shLin, stream>>>(h, Wr_l, br_l, xr, NN, HCC);

        hipMemsetAsync(smax, 0, (size_t)NN * HH * 4, stream);   // key floor
        hipMemsetAsync(den,  0, (size_t)NN * HH * 4, stream);
        hipMemsetAsync(agg,  0, (size_t)NN * HCC * 4, stream);
        hipMemsetAsync(red,  0, 2 * sizeof(double), stream);

        edge_score_kernel<<<gEdgeW, blk, 0, stream>>>(srcE, dstE, edge_attr, We_l,
                                                      xl, xr, at_l, score, smax);
        edge_den_kernel<<<gEdgeT, blk, 0, stream>>>(dstE, score, smax, den);
        edge_agg_kernel<<<gEdgeW, blk, 0, stream>>>(srcE, dstE, score, smax, den, xl, agg);

        // + conv_b, pyg graph-LayerNorm (global stats), leaky 0.01
        bias_stats_kernel<<<gElem, blk, 0, stream>>>(agg, cb_l, red, totalHC);
        pln_apply_kernel<<<gElem, blk, 0, stream>>>(agg, pw_l, pb_l, red, totalHC);

        // proj (N x 128 @ 128 x 32) + LN(32) + leaky + JK max, W staged in LDS
        gemm_wmma_f32<2, HCC, true><<<gGemm, blk, shProj, stream>>>(agg, pW_l, pB_l, raw, NN, FF);
        ln_rows_kernel<<<gRows, blk, 0, stream>>>(raw, l2w, l2b, h, jk, 1, NN);
    }
}